// ProbabilityRNN_55963423867244
// MI455X (gfx1250) — compile-verified
//
#include <hip/hip_runtime.h>

// ProbabilityRNN: per-row scalar recurrence r_{t+1} = F(s_t, r_t); F = one
// H=512 gate sweep + dot(fc_w) + sigmoid. Latency-bound: 511 sequential steps,
// 256 independent rows -> one wave per row, zero barriers, all weights hoisted
// to VGPRs. Cross-lane reduction: V_WMMA_F32_16X16X4_F32 (A=ones folds lane l
// with l^16, exact f32) + 4 DPP folds (quad_perm / row_xmask) -- no LDS round
// trips on the critical path. Gates use hw v_tanh_f32:
//   sigmoid(x) = 0.5 + 0.5*tanh(x/2)  (x0.5 pre-folded into constants).

typedef float v2f __attribute__((ext_vector_type(2)));
typedef float v8f __attribute__((ext_vector_type(8)));

#define HDIM 512
#define UPL  16   // hidden units per lane = HDIM / 32

template <int CTRL>
__device__ __forceinline__ float dpp_fold_add(float v) {
    // v += lane-permuted(v); CTRL: 0xB1 xor1, 0x4E xor2, 0x164 xor4, 0x168 xor8
    int p = __builtin_amdgcn_update_dpp(0, __float_as_int(v), CTRL, 0xF, 0xF, true);
    return v + __int_as_float(p);
}

__global__ void __launch_bounds__(64, 1)
prnn_scan_kernel(const float* __restrict__ s,
                 const float* __restrict__ W_ih,
                 const float* __restrict__ b_ih,
                 const float* __restrict__ b_hh,
                 const float* __restrict__ fc_w,
                 const float* __restrict__ fc_b,
                 float* __restrict__ out,
                 int B, int T)
{
    const int wavesPerBlock = blockDim.x >> 5;
    const int row  = blockIdx.x * wavesPerBlock + (threadIdx.x >> 5);
    const int lane = threadIdx.x & 31;
    if (row >= B) return;              // uniform per wave

    // ---- hoist per-unit constants (loop-invariant, s in {0,1} pre-folded) ----
    // r/z gates evaluated as 0.5+0.5*tanh(arg/2): store 0.5x-scaled consts.
    // n gate: n = tanh(inn + rg*b_hn): full-scale consts.
    float Wr1h[UPL], Cr0h[UPL], Cr1h[UPL];
    float Wz1h[UPL], Cz0h[UPL], Cz1h[UPL];
    float Wn1 [UPL], Cn0 [UPL], Cn1 [UPL];
    float Bh  [UPL], Fw  [UPL];

#pragma unroll
    for (int i = 0; i < UPL; ++i) {
        int u = i * 32 + lane;                      // coalesced across lanes
        float wr0 = W_ih[2*u + 0],            wr1 = W_ih[2*u + 1];
        float wz0 = W_ih[2*(HDIM + u) + 0],   wz1 = W_ih[2*(HDIM + u) + 1];
        float wn0 = W_ih[2*(2*HDIM + u) + 0], wn1 = W_ih[2*(2*HDIM + u) + 1];
        float cr  = b_ih[u]        + b_hh[u];
        float cz  = b_ih[HDIM + u] + b_hh[HDIM + u];
        float cn  = b_ih[2*HDIM + u];
        Wr1h[i] = 0.5f * wr1;  Cr0h[i] = 0.5f * cr;  Cr1h[i] = 0.5f * (cr + wr0);
        Wz1h[i] = 0.5f * wz1;  Cz0h[i] = 0.5f * cz;  Cz1h[i] = 0.5f * (cz + wz0);
        Wn1 [i] = wn1;         Cn0 [i] = cn;         Cn1 [i] = cn + wn0;
        Bh  [i] = b_hh[2*HDIM + u];
        Fw  [i] = fc_w[u];
    }
    const float fcbh = 0.5f * fc_b[0];

    const float* srow = s   + (size_t)row * T;
    float*       orow = out + (size_t)row * (T - 1);

    const v2f a_ones = {1.0f, 1.0f};               // WMMA A = ones(16x4)
    const v8f c_zero = {};

    float r  = 0.55f;                              // P0
    float sv = srow[0];

    // one H-sweep, specialized on the wave-uniform s bit (no cndmasks)
#define SWEEP(CRH, CZH, CNF)                                                    \
    _Pragma("unroll")                                                           \
    for (int i = 0; i < UPL; ++i) {                                             \
        float rg = fmaf(__builtin_amdgcn_tanhf(fmaf(r, Wr1h[i], CRH[i])),       \
                        0.5f, 0.5f);                          /* reset gate  */ \
        float zg = fmaf(__builtin_amdgcn_tanhf(fmaf(r, Wz1h[i], CZH[i])),       \
                        0.5f, 0.5f);                          /* update gate */ \
        float nv = __builtin_amdgcn_tanhf(fmaf(rg, Bh[i],                       \
                                               fmaf(r, Wn1[i], CNF[i])));       \
        acc = fmaf(fmaf(-zg, nv, nv), Fw[i], acc);            /* (1-z)n.fw  */  \
    }

    for (int t = 0; t < T - 1; ++t) {
        const bool s1 = sv > 0.5f;                 // wave-uniform
        float svn = srow[t + 1];                   // prefetch (t+1 <= T-1)

        float acc = 0.0f;
        if (s1) { SWEEP(Cr1h, Cz1h, Cn1) } else { SWEEP(Cr0h, Cz0h, Cn0) }

        // ---- 32-lane reduction: WMMA folds l with l^16, DPP finishes ----
        v2f bm; bm.x = acc; bm.y = 0.0f;
        v8f d = __builtin_amdgcn_wmma_f32_16x16x4_f32(
                    false, a_ones, false, bm, (short)0, c_zero, false, false);
        float q = d[0];                            // lane l: p_(l%16)+p_(l%16+16)
        q = dpp_fold_add<0xB1>(q);                 // xor1  (quad_perm 1,0,3,2)
        q = dpp_fold_add<0x4E>(q);                 // xor2  (quad_perm 2,3,0,1)
        q = dpp_fold_add<0x164>(q);                // xor4  (row_xmask 4)
        q = dpp_fold_add<0x168>(q);                // xor8  (row_xmask 8)

        // r_new = sigmoid(q + fc_b) = 0.5 + 0.5*tanh((q + fc_b)/2)
        r = fmaf(__builtin_amdgcn_tanhf(fmaf(q, 0.5f, fcbh)), 0.5f, 0.5f);
        if (lane == 0) orow[t] = r;
        sv = svn;
    }
#undef SWEEP
}

extern "C" void kernel_launch(void* const* d_in, const int* in_sizes, int n_in,
                              void* d_out, int out_size, void* d_ws, size_t ws_size,
                              hipStream_t stream) {
    const float* s    = (const float*)d_in[0];
    // d_in[1] = lengths (unused by reference), d_in[3] = W_hh (inert: h_prev==0)
    const float* W_ih = (const float*)d_in[2];
    const float* b_ih = (const float*)d_in[4];
    const float* b_hh = (const float*)d_in[5];
    const float* fc_w = (const float*)d_in[6];
    const float* fc_b = (const float*)d_in[7];
    float* out = (float*)d_out;

    const int B = in_sizes[1];           // lengths has one entry per row
    const int T = in_sizes[0] / B;

    const int threads = 64;              // 2 waves per block, 1 row per wave
    const int rowsPerBlock = threads / 32;
    const int blocks = (B + rowsPerBlock - 1) / rowsPerBlock;
    prnn_scan_kernel<<<blocks, threads, 0, stream>>>(s, W_ih, b_ih, b_hh,
                                                     fc_w, fc_b, out, B, T);
}